// SnakeFastLoss_59871844106878
// MI455X (gfx1250) — compile-verified
//
#include <hip/hip_runtime.h>
#include <stdint.h>

// ---------------- constants from the reference ----------------
#define STEPSZ    0.1f
#define ALPHA_C   0.01f
#define BETA_C    0.01f
#define NSTEPS    20
#define DMAX_C    15.0f
#define EXTGRADF  10.0f
#define BATCH     8
#define IMH       512
#define IMW       512
#define NNODE     256
#define KRAD      6          // filter std=2 -> r = 6, K = 13
#define KSZ       13
#define NCOL      16         // 8 batches x 2 coords packed as WMMA columns
#define LDP       264        // padded row length for LDS rhs/pos buffers

typedef float v2f __attribute__((ext_vector_type(2)));
typedef float v8f __attribute__((ext_vector_type(8)));

// =====================================================================
// Kernel 1: gradient image  gimg[b,ch,y,x] = 10 * (fltr[ch] (*) pred)[y,x]
// Direct 13x13 correlation from a 28x28 LDS tile (zero-padded SAME).
// =====================================================================
__global__ __launch_bounds__(256) void grad_conv_kernel(const float* __restrict__ pred,
                                                        float* __restrict__ gimg) {
    __shared__ float tile[28][28];
    const int b  = blockIdx.z;
    const int tx = threadIdx.x, ty = threadIdx.y;
    const int x0 = blockIdx.x * 16, y0 = blockIdx.y * 16;
    const float* img = pred + (size_t)b * IMH * IMW;

    const int t = ty * 16 + tx;
    for (int idx = t; idx < 28 * 28; idx += 256) {
        int r = idx / 28, c = idx % 28;
        int gy = y0 + r - KRAD, gx = x0 + c - KRAD;
        float v = 0.0f;
        if (gy >= 0 && gy < IMH && gx >= 0 && gx < IMW) v = img[gy * IMW + gx];
        tile[r][c] = v;
    }
    __syncthreads();

    // Gaussian + derivative taps (std = 2), normalized like the reference
    float g[KSZ], dg[KSZ];
    float sum = 0.0f;
#pragma unroll
    for (int k = 0; k < KSZ; ++k) {
        float xx = (float)(k - KRAD);
        float gv = expf(-xx * xx * (1.0f / 8.0f));
        g[k] = gv; sum += gv;
    }
    const float inv = 1.0f / sum;
#pragma unroll
    for (int k = 0; k < KSZ; ++k) {
        g[k] *= inv;
        dg[k] = (-(float)(k - KRAD) * 0.25f) * g[k];
    }

    float acc0 = 0.0f, acc1 = 0.0f;   // d/drow (dg_i*g_j), d/dcol (g_i*dg_j)
#pragma unroll
    for (int i = 0; i < KSZ; ++i) {
        float rs_g = 0.0f, rs_dg = 0.0f;
#pragma unroll
        for (int j = 0; j < KSZ; ++j) {
            float v = tile[ty + i][tx + j];
            rs_g  = fmaf(g[j],  v, rs_g);
            rs_dg = fmaf(dg[j], v, rs_dg);
        }
        acc0 = fmaf(dg[i], rs_g,  acc0);
        acc1 = fmaf(g[i],  rs_dg, acc1);
    }

    const int gy = y0 + ty, gx = x0 + tx;
    const size_t base = (size_t)b * 2 * IMH * IMW;
    gimg[base + (size_t)gy * IMW + gx]                       = acc0 * EXTGRADF;
    gimg[base + (size_t)IMH * IMW + (size_t)gy * IMW + gx]   = acc1 * EXTGRADF;
}

// =====================================================================
// Kernel 2: build A = I + t*(alpha*D + beta*D^2) (pentadiagonal, diag-
// dominant) and Gauss-Jordan invert in 256KB LDS. 1024 threads:
// thread (row i, quarter q) owns 64 columns of row i.
// =====================================================================
__device__ __forceinline__ float d_elem(int i, int j) {
    if (i == j) return (i == 0 || i == NNODE - 1) ? 1.0f : 2.0f;
    if (j == i + 1 || j == i - 1) return -1.0f;
    return 0.0f;
}

__global__ __launch_bounds__(1024) void invert_kernel(float* __restrict__ M) {
    extern __shared__ float A[];            // [256*256] = 256KB of the 320KB LDS
    const int t  = threadIdx.x;
    const int i  = t >> 2;
    const int q  = t & 3;
    const int c0 = q * 64;

    for (int j = c0; j < c0 + 64; ++j) {
        float d2 = 0.0f;
        for (int k = j - 1; k <= j + 1; ++k)
            if (k >= 0 && k < NNODE) d2 += d_elem(i, k) * d_elem(k, j);
        float a = STEPSZ * (ALPHA_C * d_elem(i, j) + BETA_C * d2);
        if (i == j) a += 1.0f;
        A[i * NNODE + j] = a;
    }
    __syncthreads();

    for (int k = 0; k < NNODE; ++k) {
        if (i == k) {
            // all 4 sibling lanes (same wave) read the pivot before any store
            float invp = 1.0f / A[k * NNODE + k];
            for (int j = c0; j < c0 + 64; ++j) A[k * NNODE + j] *= invp;
            if (k >= c0 && k < c0 + 64) A[k * NNODE + k] = invp;
        }
        __syncthreads();
        if (i != k) {
            float f  = A[i * NNODE + k];
            float pk = A[k * NNODE + k];           // == 1/pivot
            for (int j = c0; j < c0 + 64; ++j)
                A[i * NNODE + j] = fmaf(-f, A[k * NNODE + j], A[i * NNODE + j]);
            if (k >= c0 && k < c0 + 64) A[i * NNODE + k] = -f * pk;
        }
        __syncthreads();
    }

    for (int j = c0; j < c0 + 64; ++j) M[i * NNODE + j] = A[i * NNODE + j];
}

// =====================================================================
// Kernel 3: 20 snake steps. One workgroup, 512 threads = 16 waves.
// All 8 batches x 2 coords packed as 16 RHS columns -> each step is a
// single [256x256]x[256x16] GEMM done with V_WMMA_F32_16X16X4_F32,
// wave w computing the 16x16 tile of output rows [16w,16w+16).
// =====================================================================
__device__ __forceinline__ float bilerp512(const float* __restrict__ img, float r, float c) {
    r = fminf(fmaxf(r, 0.0f), 511.0f);
    c = fminf(fmaxf(c, 0.0f), 511.0f);
    float rf = floorf(r), cf = floorf(c);
    int r0 = (int)rf, c0 = (int)cf;
    int r1 = r0 + 1; if (r1 > 511) r1 = 511;
    int c1 = c0 + 1; if (c1 > 511) c1 = 511;
    float fr = r - rf, fc = c - cf;
    float v00 = img[r0 * IMW + c0], v10 = img[r1 * IMW + c0];
    float v01 = img[r0 * IMW + c1], v11 = img[r1 * IMW + c1];
    return v00 * (1.0f - fr) * (1.0f - fc) + v10 * fr * (1.0f - fc)
         + v01 * (1.0f - fr) * fc          + v11 * fr * fc;
}

__global__ __launch_bounds__(512) void snake_kernel(const float* __restrict__ gimg,
                                                    const float* __restrict__ node_pos,
                                                    const float* __restrict__ M,
                                                    float* __restrict__ pos_out) {
    // column-major (transposed) buffers: [col = b*2+d][row n], padded rows
    __shared__ float posT[NCOL][LDP];
    __shared__ float rhsT[NCOL][LDP];

    const int t     = threadIdx.x;
    const int wave  = t >> 5;
    const int lane  = t & 31;
    const int lmod  = lane & 15;    // M-row-in-tile (A), column n (B/C/D)
    const int lhalf = lane >> 4;    // K sub-offset / row-half select

    for (int idx = t; idx < BATCH * NNODE; idx += 512) {
        int b = idx >> 8, n = idx & 255;
        posT[b * 2 + 0][n] = node_pos[(size_t)(b * NNODE + n) * 2 + 0];
        posT[b * 2 + 1][n] = node_pos[(size_t)(b * NNODE + n) * 2 + 1];
    }
    __syncthreads();

    const int   m0   = wave * 16;
    const float* Arow = M + (size_t)(m0 + lmod) * NNODE + lhalf * 2;

    for (int step = 0; step < NSTEPS; ++step) {
        // ---- external force + rhs = pos + STEPSZ * f ----
        for (int idx = t; idx < BATCH * NNODE; idx += 512) {
            int b = idx >> 8, n = idx & 255;
            float r = posT[b * 2 + 0][n];
            float c = posT[b * 2 + 1][n];
            const float* g0 = gimg + (size_t)(b * 2 + 0) * IMH * IMW;
            const float* g1 = gimg + (size_t)(b * 2 + 1) * IMH * IMW;
            rhsT[b * 2 + 0][n] = fmaf(STEPSZ, bilerp512(g0, r, c), r);
            rhsT[b * 2 + 1][n] = fmaf(STEPSZ, bilerp512(g1, r, c), c);
        }
        __syncthreads();

        // ---- implicit step: pos_new = M @ rhs  (WMMA f32 16x16x4) ----
        v8f acc = {0.f, 0.f, 0.f, 0.f, 0.f, 0.f, 0.f, 0.f};
        const float* Bcol = &rhsT[lmod][lhalf * 2];
#pragma unroll 4
        for (int k = 0; k < NNODE; k += 4) {
            v2f a  = *(const v2f*)(Arow + k);   // A[m][k + lhalf*2 .. +1]
            v2f bb = *(const v2f*)(Bcol + k);   // rhs[k + lhalf*2 .. +1][n]
            acc = __builtin_amdgcn_wmma_f32_16x16x4_f32(
                false, a, false, bb, (short)0, acc, false, false);
        }

        // ---- clip to [0, H-1 / W-1] and write back transposed ----
        const float bound = (lmod & 1) ? (float)(IMW - 1) : (float)(IMH - 1);
#pragma unroll
        for (int j = 0; j < 8; ++j) {
            int mrow = m0 + j + lhalf * 8;      // C/D layout: lanes 16-31 -> M+8
            float v = fminf(fmaxf(acc[j], 0.0f), bound);
            posT[lmod][mrow] = v;
        }
        __syncthreads();
    }

    for (int idx = t; idx < BATCH * NNODE; idx += 512) {
        int b = idx >> 8, n = idx & 255;
        pos_out[(size_t)(b * NNODE + n) * 2 + 0] = posT[b * 2 + 0][n];
        pos_out[(size_t)(b * NNODE + n) * 2 + 1] = posT[b * 2 + 1][n];
    }
}

// =====================================================================
// Kernel 4: distance-map render + squared-error partial sums.
// One block per (batch,row). Node data staged to LDS with CDNA5
// global_load_async_to_lds_b128 (+ s_wait_asynccnt).
// =====================================================================
__global__ __launch_bounds__(256) void render_loss_kernel(const float* __restrict__ pred,
                                                          const float* __restrict__ pos,
                                                          const float* __restrict__ widths,
                                                          float* __restrict__ partials) {
    __shared__ __align__(16) float nodes_rc[NNODE * 2];  // interleaved (r,c)
    __shared__ __align__(16) float nw[NNODE];
    __shared__ float red[256];

    const int b = blockIdx.x >> 9;
    const int y = blockIdx.x & 511;
    const int t = threadIdx.x;

    // async stage: 128 lanes x 16B of pos, 64 lanes x 16B of widths
    if (t < 128) {
        uint32_t lds = (uint32_t)(uintptr_t)(&nodes_rc[t * 4]);
        uint64_t ga  = (uint64_t)(uintptr_t)(pos + (size_t)b * NNODE * 2 + t * 4);
        asm volatile("global_load_async_to_lds_b128 %0, %1, off" :: "v"(lds), "v"(ga) : "memory");
    } else if (t < 192) {
        uint32_t lds = (uint32_t)(uintptr_t)(&nw[(t - 128) * 4]);
        uint64_t ga  = (uint64_t)(uintptr_t)(widths + (size_t)b * NNODE + (t - 128) * 4);
        asm volatile("global_load_async_to_lds_b128 %0, %1, off" :: "v"(lds), "v"(ga) : "memory");
    }
    asm volatile("s_wait_asynccnt 0" ::: "memory");
    __syncthreads();

    const float fy = (float)y;
    const float fx0 = (float)t, fx1 = (float)(t + 256);
    float md0 = 1e30f, md1 = 1e30f;
#pragma unroll 4
    for (int n = 0; n < NNODE; ++n) {
        float dy  = fy - nodes_rc[2 * n];
        float pc  = nodes_rc[2 * n + 1];
        float w   = nw[n];
        float dy2 = dy * dy;
        float dx0 = fx0 - pc;
        float dx1 = fx1 - pc;
        md0 = fminf(md0, sqrtf(fmaf(dx0, dx0, dy2)) - w);
        md1 = fminf(md1, sqrtf(fmaf(dx1, dx1, dy2)) - w);
    }
    float dm0 = fminf(fmaxf(md0, 0.0f), DMAX_C);
    float dm1 = fminf(fmaxf(md1, 0.0f), DMAX_C);

    const float* prow = pred + ((size_t)b * IMH + y) * IMW;
    float e0 = prow[t] - dm0;
    float e1 = prow[t + 256] - dm1;

    red[t] = e0 * e0 + e1 * e1;
    __syncthreads();
    for (int off = 128; off > 0; off >>= 1) {
        if (t < off) red[t] += red[t + off];
        __syncthreads();
    }
    if (t == 0) partials[blockIdx.x] = red[0];
}

// =====================================================================
// Kernel 5: deterministic final reduction -> mean
// =====================================================================
__global__ __launch_bounds__(256) void final_reduce_kernel(const float* __restrict__ partials,
                                                           float* __restrict__ out) {
    __shared__ float red[256];
    const int t = threadIdx.x;
    float s = 0.0f;
    for (int i = t; i < BATCH * IMH; i += 256) s += partials[i];
    red[t] = s;
    __syncthreads();
    for (int off = 128; off > 0; off >>= 1) {
        if (t < off) red[t] += red[t + off];
        __syncthreads();
    }
    if (t == 0) out[0] = red[0] * (1.0f / (float)(BATCH * IMH * IMW));
}

// =====================================================================
extern "C" void kernel_launch(void* const* d_in, const int* in_sizes, int n_in,
                              void* d_out, int out_size, void* d_ws, size_t ws_size,
                              hipStream_t stream) {
    (void)in_sizes; (void)n_in; (void)out_size; (void)ws_size;
    const float* pred     = (const float*)d_in[0];   // [8,1,512,512]
    const float* node_pos = (const float*)d_in[1];   // [8,256,2]
    const float* widths   = (const float*)d_in[2];   // [8,256]
    float* out = (float*)d_out;

    float* ws       = (float*)d_ws;
    float* M        = ws;              // 65536 floats
    float* pos_out  = ws + 65536;      // 4096 floats
    float* partials = ws + 69632;      // 4096 floats
    float* gimg     = ws + 73728;      // 8*2*512*512 floats

    dim3 cblk(16, 16, 1), cgrd(IMW / 16, IMH / 16, BATCH);
    grad_conv_kernel<<<cgrd, cblk, 0, stream>>>(pred, gimg);

    invert_kernel<<<1, 1024, NNODE * NNODE * sizeof(float), stream>>>(M);

    snake_kernel<<<1, 512, 0, stream>>>(gimg, node_pos, M, pos_out);

    render_loss_kernel<<<BATCH * IMH, 256, 0, stream>>>(pred, pos_out, widths, partials);

    final_reduce_kernel<<<1, 256, 0, stream>>>(partials, out);
}